// DigitCapsules_70171175682299
// MI455X (gfx1250) — compile-verified
//
#include <hip/hip_runtime.h>

// ---------------------------------------------------------------------------
// Capsule dynamic routing, fused for MI455X (gfx1250, wave32, WMMA).
//
// u_hat[b,p,c,o] = sum_i W[p,c,o,i] * u[b,p,i]  -> per p, a [16b x 8] x [8 x 16o]
// GEMM tile per capsule c, done with two V_WMMA_F32_16X16X4_F32 (K=4 each).
// u_hat is never written to HBM: each routing iteration recomputes it from
// L2-resident u (16.8MB) and W (10.5MB). Only b_logits (21MB) + s (160KB)
// live in the workspace; v lives in d_out.
// ---------------------------------------------------------------------------

typedef __attribute__((ext_vector_type(2))) float v2f;
typedef __attribute__((ext_vector_type(8))) float v8f;

#define B_    256
#define P_    2048
#define C_    10
#define OUT_  16
#define IN_   8
#define BT    16            // batch tile (WMMA M)
#define WAVES 8             // waves per workgroup
#define PPW   16            // primary capsules per wave
#define PP    (WAVES * PPW) // 128 p per workgroup
#define NPCH  (P_ / PP)     // 16 p-chunks
#define NBT   (B_ / BT)     // 16 batch tiles

// One routing step: [t>0: bl += <u_hat, v_prev>; cc = softmax(bl)]
//                   [t==0: cc = 0.1]; s += cc * u_hat  (atomic reduce)
__global__ __launch_bounds__(256) void caps_route(
    const float* __restrict__ u,      // [B,P,IN]
    const float* __restrict__ W,      // [P,C,OUT,IN]
    float* __restrict__ bl,           // [B,P,C] routing logits
    float* __restrict__ s,            // [B,C,OUT] accumulator (pre-zeroed)
    const float* __restrict__ vprev,  // [B,C,OUT] previous v (d_out)
    int t)
{
  __shared__ float s_s[BT * C_ * OUT_];      // per-workgroup s partials
  __shared__ float v_s[BT * C_ * OUT_];      // v tile for this batch tile
  __shared__ float cc_s[WAVES][BT][C_];      // softmax coefficients per wave
  __shared__ float bi_s[WAVES][BT][C_];      // agreement increments per wave

  const int tid  = threadIdx.x;
  const int w    = tid >> 5;
  const int lane = tid & 31;
  const int hl   = lane & 15;   // o index / b-row within half
  const int hi   = lane >> 4;   // half-wave select
  const int btile = blockIdx.y * BT;
  const int p0    = blockIdx.x * PP + w * PPW;

  for (int i = tid; i < BT * C_ * OUT_; i += 256) s_s[i] = 0.f;
  if (t > 0) {
    for (int i = tid; i < BT * C_ * OUT_; i += 256) {
      const int m = i / (C_ * OUT_);
      v_s[i] = vprev[(size_t)(btile + m) * C_ * OUT_ + (i % (C_ * OUT_))];
    }
  }
  __syncthreads();

  // Per-wave s accumulator mirroring WMMA C/D layout:
  // sacc[c][r] in lane l -> s[btile + r + 8*(l>=16)][c][l%16]
  float sacc[C_][8];
#pragma unroll
  for (int c = 0; c < C_; ++c) {
#pragma unroll
    for (int r = 0; r < 8; ++r) sacc[c][r] = 0.f;
  }

  // A-matrix rows: lane hl owns batch row (btile+hl); K = 2*hi + {0,1} (+4)
  const float* ub = u + (size_t)(btile + hl) * P_ * IN_;

  for (int pp = 0; pp < PPW; ++pp) {
    const int p = p0 + pp;

    // ---- A operands (u rows), 16x4 f32 layout -------------------------
    const float* up = ub + (size_t)p * IN_;
    v2f a0, a1;
    a0.x = up[2 * hi + 0]; a0.y = up[2 * hi + 1];   // K-tile 0: i = 0..3
    a1.x = up[2 * hi + 4]; a1.y = up[2 * hi + 5];   // K-tile 1: i = 4..7

    // ---- u_hat tile: 10 capsules x (16b x 16o), 2 WMMAs each ----------
    // B operand 4x16 f32: lane hl = N(o), VGPR j + 2*hi = K(i)
    const float* Wp = W + (size_t)p * C_ * OUT_ * IN_ + hl * IN_ + 2 * hi;
    v8f uh[C_];
#pragma unroll
    for (int c = 0; c < C_; ++c) {
      const float* wpc = Wp + c * OUT_ * IN_;
      v2f b0, b1;
      b0.x = wpc[0]; b0.y = wpc[1];
      b1.x = wpc[4]; b1.y = wpc[5];
      v8f acc = {0.f, 0.f, 0.f, 0.f, 0.f, 0.f, 0.f, 0.f};
      acc = __builtin_amdgcn_wmma_f32_16x16x4_f32(false, a0, false, b0,
                                                  (short)0, acc, false, false);
      acc = __builtin_amdgcn_wmma_f32_16x16x4_f32(false, a1, false, b1,
                                                  (short)0, acc, false, false);
      uh[c] = acc;
    }

    if (t > 0) {
      // ---- agreement: bi[m][c] = sum_o u_hat[m,c,o] * v[m,c,o] --------
#pragma unroll
      for (int c = 0; c < C_; ++c) {
#pragma unroll
        for (int r = 0; r < 8; ++r) {
          const int m = r + 8 * hi;
          float d = uh[c][r] * v_s[(m * C_ + c) * OUT_ + hl];
          d += __shfl_xor(d, 1, 32);
          d += __shfl_xor(d, 2, 32);
          d += __shfl_xor(d, 4, 32);
          d += __shfl_xor(d, 8, 32);   // reduced over 16 o-lanes
          if (hl == 0) bi_s[w][m][c] = d;
        }
      }
      __builtin_amdgcn_wave_barrier();

      // ---- logit update + softmax over C, one lane per batch row ------
      if (hi == 0) {
        const int m = hl;
        float* blrow = bl + ((size_t)(btile + m) * P_ + p) * C_;
        float row[C_], e[C_];
        float mx = -3.0e38f;
#pragma unroll
        for (int c = 0; c < C_; ++c) {
          row[c] = blrow[c] + bi_s[w][m][c];
          mx = fmaxf(mx, row[c]);
        }
        float sum = 0.f;
#pragma unroll
        for (int c = 0; c < C_; ++c) { e[c] = __expf(row[c] - mx); sum += e[c]; }
        const float inv = 1.f / sum;
#pragma unroll
        for (int c = 0; c < C_; ++c) {
          cc_s[w][m][c] = e[c] * inv;
          if (t < 2) blrow[c] = row[c];   // last iteration's update is unused
        }
      }
      __builtin_amdgcn_wave_barrier();
    } else {
      // softmax of zero logits is exactly uniform 1/C
      if (hi == 0) {
#pragma unroll
        for (int c = 0; c < C_; ++c) cc_s[w][hl][c] = 0.1f;
      }
      __builtin_amdgcn_wave_barrier();
    }

    // ---- s partial accumulation (per-b coefficient broadcast) ---------
#pragma unroll
    for (int c = 0; c < C_; ++c) {
#pragma unroll
      for (int r = 0; r < 8; ++r)
        sacc[c][r] += cc_s[w][r + 8 * hi][c] * uh[c][r];
    }
    __builtin_amdgcn_wave_barrier();
  }

  // ---- cross-wave reduce via LDS atomics, then one global atomic pass --
#pragma unroll
  for (int c = 0; c < C_; ++c) {
#pragma unroll
    for (int r = 0; r < 8; ++r)
      atomicAdd(&s_s[((r + 8 * hi) * C_ + c) * OUT_ + hl], sacc[c][r]);
  }
  __syncthreads();
  for (int i = tid; i < BT * C_ * OUT_; i += 256)
    atomicAdd(&s[(size_t)btile * C_ * OUT_ + i], s_s[i]);
}

// v = squash(s) per (b,c) 16-vector; also resets s for the next iteration.
__global__ __launch_bounds__(160) void caps_squash(float* __restrict__ s,
                                                   float* __restrict__ vout)
{
  const int b = blockIdx.x;
  const size_t idx = (size_t)b * C_ * OUT_ + threadIdx.x;  // tid = c*16 + o
  const float val = s[idx];
  float sq = val * val;
  sq += __shfl_xor(sq, 1, 32);
  sq += __shfl_xor(sq, 2, 32);
  sq += __shfl_xor(sq, 4, 32);
  sq += __shfl_xor(sq, 8, 32);          // sum over the 16 o-lanes
  const float scale = (sq / (1.f + sq)) * rsqrtf(sq + 1e-9f);
  vout[idx] = scale * val;
  s[idx] = 0.f;
}

extern "C" void kernel_launch(void* const* d_in, const int* in_sizes, int n_in,
                              void* d_out, int out_size, void* d_ws, size_t ws_size,
                              hipStream_t stream) {
  const float* u = (const float*)d_in[0];   // [B,P,IN]
  const float* W = (const float*)d_in[1];   // [1,P,C,OUT,IN]
  float* out = (float*)d_out;               // [B,C,OUT]
  float* bl  = (float*)d_ws;                // [B,P,C]
  float* s   = bl + (size_t)B_ * P_ * C_;   // [B,C,OUT]

  // Deterministic re-init every call (bl and s are contiguous in ws).
  hipMemsetAsync(d_ws, 0,
                 ((size_t)B_ * P_ * C_ + (size_t)B_ * C_ * OUT_) * sizeof(float),
                 stream);

  const dim3 grid(NPCH, NBT), block(256);
  for (int t = 0; t < 3; ++t) {
    caps_route<<<grid, block, 0, stream>>>(u, W, bl, s, out, t);
    caps_squash<<<dim3(B_), dim3(160), 0, stream>>>(s, out);  // final write -> d_out
  }
}